// Mean_Filter_68719477233
// MI455X (gfx1250) — compile-verified
//
#include <hip/hip_runtime.h>

// MI455X 5x5 edge-clipped box mean, (32,3,512,512) f32.
// Separable: horizontal 5-sum in LDS, vertical 5-sum as banded matmul on the
// WMMA pipe (V_WMMA_F32_16X16X4_F32). The edge-clip divisor cnt = xcnt*ycnt is
// separable, so 1/ycnt is folded into the band A-matrix (per output row) and
// 1/xcnt into B (per output column) -> WMMA emits the final mean directly.
// Interior tiles stage the halo via GLOBAL_LOAD_ASYNC_TO_LDS_B128 and stream
// results out via GLOBAL_STORE_ASYNC_FROM_LDS_B128 (ASYNCcnt-tracked).
// Bandwidth-bound: ~200 MB @ 23.3 TB/s => ~8.6 us floor.

typedef float v2f __attribute__((ext_vector_type(2)));
typedef float v8f __attribute__((ext_vector_type(8)));

// Async-LDS builtins take pointer-to-int4: global (AS1) side, LDS (AS3) side.
typedef int v4i __attribute__((__vector_size__(16)));
typedef __attribute__((address_space(1))) v4i* g4p;
typedef __attribute__((address_space(3))) v4i* l4p;

#define IMG_H 512
#define IMG_W 512
#define TILE 64
#define IN_ROWS 68       // gy0-2 .. gy0+65
#define IN_COLS 72       // col origin gx0-4 (16B-aligned), gx0-4 .. gx0+67
#define IN_STRIDE 72     // 288 B rows: 16B-aligned for b128 LDS traffic
#define HS_STRIDE 80     // bank-aware stride for horizontal sums
#define NTHREADS 256

#if __has_builtin(__builtin_amdgcn_global_load_async_to_lds_b128)
#define HAVE_ASYNC_LD 1
#else
#define HAVE_ASYNC_LD 0
#endif
#if __has_builtin(__builtin_amdgcn_global_store_async_from_lds_b128)
#define HAVE_ASYNC_ST 1
#else
#define HAVE_ASYNC_ST 0
#endif

__device__ __forceinline__ void wait_async0() {
#if __has_builtin(__builtin_amdgcn_s_wait_asynccnt)
    __builtin_amdgcn_s_wait_asynccnt(0);
#else
    asm volatile("s_wait_asynccnt 0" ::: "memory");
#endif
}

__device__ __forceinline__ float fast_rcp(float v) {
#if __has_builtin(__builtin_amdgcn_rcpf)
    return __builtin_amdgcn_rcpf(v);
#else
    return 1.0f / v;
#endif
}

__global__ __launch_bounds__(NTHREADS)
void Mean_Filter_68719477233_kernel(const float* __restrict__ x,
                                    float* __restrict__ out) {
    __shared__ float s_in[IN_ROWS * IN_STRIDE];  // 68 x 72 = 19,584 B (reused for output staging)
    __shared__ float s_hs[IN_ROWS * HS_STRIDE];  // 68 x 80 = 21,760 B

    const int tid  = threadIdx.x;
    const int lane = tid & 31;
    const int wave = tid >> 5;

    const int tiles_per_plane = (IMG_H / TILE) * (IMG_W / TILE);  // 64
    const int plane = blockIdx.x / tiles_per_plane;
    const int t     = blockIdx.x % tiles_per_plane;
    const int gx0 = (t % (IMG_W / TILE)) * TILE;
    const int gy0 = (t / (IMG_W / TILE)) * TILE;

    const float* __restrict__ src = x   + (size_t)plane * IMG_H * IMG_W;
    float* __restrict__       dst = out + (size_t)plane * IMG_H * IMG_W;

    // ---- Stage 68x72 halo tile into LDS (col origin gx0-4, row origin gy0-2).
    const bool interior = (gx0 >= TILE) && (gx0 + TILE < IMG_W) &&
                          (gy0 >= TILE) && (gy0 + TILE < IMG_H);
#if HAVE_ASYNC_LD
    if (interior) {
        // 68 rows x 18 float4 = 1224 async b128 global->LDS copies, no VGPR return.
        for (int idx = tid; idx < IN_ROWS * (IN_COLS / 4); idx += NTHREADS) {
            const int r = idx / (IN_COLS / 4);
            const int q = idx - r * (IN_COLS / 4);
            const float* g = src + (size_t)(gy0 - 2 + r) * IMG_W + (gx0 - 4) + 4 * q;
            __builtin_amdgcn_global_load_async_to_lds_b128(
                (g4p)(void*)g, (l4p)&s_in[r * IN_STRIDE + 4 * q], 0, 0);
        }
        wait_async0();
    } else
#endif
    {
        for (int idx = tid; idx < IN_ROWS * IN_COLS; idx += NTHREADS) {
            const int r = idx / IN_COLS;
            const int c = idx - r * IN_COLS;
            const int gy = gy0 - 2 + r;
            const int gx = gx0 - 4 + c;
            float v = 0.0f;
            if (gy >= 0 && gy < IMG_H && gx >= 0 && gx < IMG_W)
                v = src[gy * IMG_W + gx];
            s_in[r * IN_STRIDE + c] = v;
        }
    }
    __syncthreads();

    // ---- Horizontal 5-tap sums: 68 rows x 64 cols (center gx0+c uses cols c+2..c+6).
    for (int idx = tid; idx < IN_ROWS * TILE; idx += NTHREADS) {
        const int r = idx / TILE;
        const int c = idx - r * TILE;
        const float* p = &s_in[r * IN_STRIDE + c + 2];
        s_hs[r * HS_STRIDE + c] = p[0] + p[1] + p[2] + p[3] + p[4];
    }
    __syncthreads();

    // ---- Vertical 5-tap + divide as banded matmul:
    // Out(16x16) = A(16x20) * B(20x16), A[m][k] = (m<=k<=m+4) ? 1/ycnt(m) : 0,
    // B[k][n] = hsum[k][n] * 1/xcnt(n). Five K=4 WMMA steps.
    const int mhalf = lane >> 4;   // 0: lanes 0-15, 1: lanes 16-31
    const int nl    = lane & 15;
    for (int tt = wave * 2; tt < wave * 2 + 2; ++tt) {
        const int tx  = tt & 3;
        const int ty  = tt >> 2;
        const int x0t = tx * 16;
        const int r0  = ty * 16;       // hsum row of k=0 (image row gy0+ty*16-2)

        // 1/ycnt for this lane's A-row (m = nl), 1/xcnt for this lane's B-col.
        const int gym = gy0 + ty * 16 + nl;
        int yl = gym - 2; if (yl < 0) yl = 0;
        int yr = gym + 2; if (yr > IMG_H - 1) yr = IMG_H - 1;
        const float ry = fast_rcp((float)(yr - yl + 1));

        const int gxc = gx0 + x0t + nl;
        int xl = gxc - 2; if (xl < 0) xl = 0;
        int xr = gxc + 2; if (xr > IMG_W - 1) xr = IMG_W - 1;
        const float rx = fast_rcp((float)(xr - xl + 1));

        v8f acc = {};
#pragma unroll
        for (int kk = 0; kk < 5; ++kk) {
            // A-layout (16x4 f32): lanes 0-15 hold K=kb,kb+1; lanes 16-31 K=kb+2,kb+3.
            const int kb = kk * 4 + mhalf * 2;
            v2f a, b;
            a.x = (kb     >= nl && kb     <= nl + 4) ? ry : 0.0f;
            a.y = (kb + 1 >= nl && kb + 1 <= nl + 4) ? ry : 0.0f;
            b.x = s_hs[(r0 + kb    ) * HS_STRIDE + x0t + nl] * rx;
            b.y = s_hs[(r0 + kb + 1) * HS_STRIDE + x0t + nl] * rx;
            acc = __builtin_amdgcn_wmma_f32_16x16x4_f32(
                false, a, false, b, (short)0, acc, false, false);
        }
        // D-layout: VGPR j -> row j (lanes 0-15) / j+8 (lanes 16-31), col nl.
        // Stage into s_in (dead after hsum pass) for vectorized stores.
#pragma unroll
        for (int j = 0; j < 8; ++j)
            s_in[(ty * 16 + j + mhalf * 8) * IN_STRIDE + x0t + nl] = acc[j];
    }
    __syncthreads();

    // ---- Stream 64x64 results out as b128: 64 rows x 16 float4.
#if HAVE_ASYNC_ST
    for (int idx = tid; idx < TILE * (TILE / 4); idx += NTHREADS) {
        const int r = idx >> 4;
        const int q = idx & 15;
        float* g = dst + (size_t)(gy0 + r) * IMG_W + gx0 + 4 * q;
        __builtin_amdgcn_global_store_async_from_lds_b128(
            (g4p)(void*)g, (l4p)&s_in[r * IN_STRIDE + 4 * q], 0, 0);
    }
    wait_async0();
#else
    for (int idx = tid; idx < TILE * (TILE / 4); idx += NTHREADS) {
        const int r = idx >> 4;
        const int q = idx & 15;
        *(float4*)(dst + (size_t)(gy0 + r) * IMG_W + gx0 + 4 * q) =
            *(const float4*)&s_in[r * IN_STRIDE + 4 * q];
    }
#endif
}

extern "C" void kernel_launch(void* const* d_in, const int* in_sizes, int n_in,
                              void* d_out, int out_size, void* d_ws, size_t ws_size,
                              hipStream_t stream) {
    (void)n_in; (void)d_ws; (void)ws_size; (void)out_size;
    const float* x = (const float*)d_in[0];
    float* out = (float*)d_out;
    const int planes = in_sizes[0] / (IMG_H * IMG_W);            // 32*3 = 96
    const int tiles_per_plane = (IMG_H / TILE) * (IMG_W / TILE); // 64
    dim3 grid(planes * tiles_per_plane);
    dim3 block(NTHREADS);
    Mean_Filter_68719477233_kernel<<<grid, block, 0, stream>>>(x, out);
}